// BLEULoss_64166811402662
// MI455X (gfx1250) — compile-verified
//
#include <hip/hip_runtime.h>

// ---------------------------------------------------------------------------
// BLEU-1 set-precision loss, CDNA5 (gfx1250).
//
// precision_i = |set(pred_i) & set(tgt_i)| / |set(pred_i)| ; out = 1 - mean_i.
// VOCAB=32000 -> per-row set = 1000-dword LDS bitmap. O(B*(Lp+Lt)) instead of
// the reference's O(B*Lp*(Lp+Lt)). Total traffic 1 MB => DRAM time ~45ns at
// 23.3 TB/s; the kernel is latency bound, so we use CDNA5 async-to-LDS copies
// (ASYNCcnt) to overlap the token-row staging with LDS bitmap zeroing.
// ---------------------------------------------------------------------------

#define VOCAB     32000
#define BM_WORDS  ((VOCAB + 31) / 32)   // 1000 dwords = 4000 B per bitmap
#define NROWS     64
#define LP        2048
#define LT        2048
#define THREADS   256                   // 8 waves (wave32)

// ---- CDNA5 async global->LDS path (guarded; falls back to sync loads) -----
#if defined(__AMDGCN__) && defined(__gfx1250__)
#  if __has_builtin(__builtin_amdgcn_global_load_async_to_lds_b128) && \
      __has_builtin(__builtin_amdgcn_s_wait_asynccnt)
#    define USE_ASYNC_LDS 1
#  endif
#endif
#ifndef USE_ASYNC_LDS
#  define USE_ASYNC_LDS 0
#endif

#if USE_ASYNC_LDS
// Builtin signature (from hipcc diagnostic): param0 is
//   __attribute__((vector_size(16))) int __device__ *   (i.e. v4i in AS1),
// param1 the LDS-side v4i pointer (AS3), then two constant ints (offset, cpol).
typedef int v4i __attribute__((__vector_size__(16)));
typedef __attribute__((address_space(1))) v4i glb_v4i;   // global, 64-bit ptr
typedef __attribute__((address_space(3))) v4i lds_v4i;   // LDS, 32-bit ptr

// Generic flat ptr -> addrspace ptr via integer round-trip (avoids the C++
// restriction on direct addrspace pointer casts). On gfx1250 the low 32 bits
// of a flat LDS address are the LDS byte offset (ISA 10.2 aperture table),
// so the 32-bit narrowing for AS3 is value-correct.
__device__ __forceinline__ lds_v4i* to_lds(void* p) {
  return (lds_v4i*)(unsigned)(unsigned long long)p;
}
__device__ __forceinline__ glb_v4i* to_glb(const void* p) {
  return (glb_v4i*)(unsigned long long)p;
}
#endif

__global__ __launch_bounds__(THREADS)
void bleu_row_kernel(const int* __restrict__ pred,
                     const int* __restrict__ tgt,
                     float* __restrict__ row_prec) {
  __shared__ unsigned tgt_bm[BM_WORDS];
  __shared__ unsigned pred_bm[BM_WORDS];
  __shared__ unsigned s_cnt[2];               // [0]=unique, [1]=overlap
#if USE_ASYNC_LDS
  __shared__ int s_tok[LT + LP];              // staged token rows (16 KB)
#endif

  const int row = blockIdx.x;
  const int tid = threadIdx.x;
  const int* gt = tgt  + (size_t)row * LT;
  const int* gp = pred + (size_t)row * LP;

#if USE_ASYNC_LDS
  // Kick off async copies of both token rows (b128 = 4 ints per lane per op;
  // 2048 ints / (256 lanes * 4) = 2 ops per lane per row). Tracked by
  // ASYNCcnt; overlapped with the bitmap zeroing below.
  for (int base = 0; base < LT; base += THREADS * 4) {
    const int idx = base + tid * 4;
    __builtin_amdgcn_global_load_async_to_lds_b128(
        to_glb(gt + idx), to_lds(&s_tok[idx]), 0, 0);
  }
  for (int base = 0; base < LP; base += THREADS * 4) {
    const int idx = base + tid * 4;
    __builtin_amdgcn_global_load_async_to_lds_b128(
        to_glb(gp + idx), to_lds(&s_tok[LT + idx]), 0, 0);
  }
#endif

  // Zero bitmaps while the async copies are in flight.
  for (int i = tid; i < BM_WORDS; i += THREADS) {
    tgt_bm[i]  = 0u;
    pred_bm[i] = 0u;
  }
  if (tid == 0) { s_cnt[0] = 0u; s_cnt[1] = 0u; }

#if USE_ASYNC_LDS
  __builtin_amdgcn_s_wait_asynccnt(0);        // s_wait_asynccnt 0 (per-wave)
#endif
  __syncthreads();                            // all waves' copies + zeroing done

  // ---- build target-set bitmap (ds_or_b32) ----
  for (int j = tid; j < LT; j += THREADS) {
#if USE_ASYNC_LDS
    const unsigned t = (unsigned)s_tok[j];
#else
    const unsigned t = (unsigned)gt[j];
#endif
    atomicOr(&tgt_bm[t >> 5], 1u << (t & 31u));
  }
  __syncthreads();

  // ---- pred uniques + overlap (ds_or_rtn_b32 gives exact first-occurrence
  //      semantics even when duplicate tokens race within/across waves) ----
  unsigned uniq = 0u, ovl = 0u;
  for (int j = tid; j < LP; j += THREADS) {
#if USE_ASYNC_LDS
    const unsigned p = (unsigned)s_tok[LT + j];
#else
    const unsigned p = (unsigned)gp[j];
#endif
    const unsigned w = p >> 5;
    const unsigned m = 1u << (p & 31u);
    const unsigned old = atomicOr(&pred_bm[w], m);
    if (!(old & m)) {                 // this lane owns the first occurrence
      ++uniq;
      if (tgt_bm[w] & m) ++ovl;       // target bitmap is read-only by now
    }
  }
  if (uniq) atomicAdd(&s_cnt[0], uniq);       // ds_add_u32
  if (ovl)  atomicAdd(&s_cnt[1], ovl);
  __syncthreads();

  if (tid == 0) {
    const float denom = (float)s_cnt[0];
    row_prec[row] = (denom > 0.0f) ? ((float)s_cnt[1] / denom) : 0.0f;
  }
}

// Final reduce: 64 precisions -> 1 - mean. Serial accumulation in one lane
// for a fixed FP order (deterministic across replays).
__global__ __launch_bounds__(NROWS)
void bleu_finish_kernel(const float* __restrict__ row_prec,
                        float* __restrict__ out) {
  __shared__ float s[NROWS];
  const int tid = threadIdx.x;
  s[tid] = row_prec[tid];
  __syncthreads();
  if (tid == 0) {
    float sum = 0.0f;
    for (int i = 0; i < NROWS; ++i) sum += s[i];
    out[0] = 1.0f - sum * (1.0f / (float)NROWS);
  }
}

extern "C" void kernel_launch(void* const* d_in, const int* in_sizes, int n_in,
                              void* d_out, int out_size, void* d_ws, size_t ws_size,
                              hipStream_t stream) {
  const int* pred = (const int*)d_in[0];   // [64, 2048] int32
  const int* tgt  = (const int*)d_in[1];   // [64, 2048] int32
  float* row_prec = (float*)d_ws;          // 64 floats of scratch
  float* out      = (float*)d_out;         // scalar float32

  bleu_row_kernel<<<dim3(NROWS), dim3(THREADS), 0, stream>>>(pred, tgt, row_prec);
  bleu_finish_kernel<<<dim3(1), dim3(NROWS), 0, stream>>>(row_prec, out);
}